// GraphSAGE3D_5016521801769
// MI455X (gfx1250) — compile-verified
//
#include <hip/hip_runtime.h>

typedef __attribute__((ext_vector_type(2))) float v2f;
typedef __attribute__((ext_vector_type(8))) float v8f;

#define NB_B   2
#define NB_C   64
#define NB_N   50000
#define NB_K   9
#define NB_OUT 64
#define TILES_TOTAL ((NB_B * NB_N) / 16)   // 6250 (50000 % 16 == 0, tiles never straddle batch)
#define WAVES_PER_BLOCK 8
#define PSTRIDE 160   // dwords per c-pair row: 128 data + 32 pad => halves on disjoint banks

// Issue the 32 gather loads forming the B-fragments (4x16 f32 tiles, 16 c-chunks)
// for one neighbor column-tile. Lanes 0-15 carry K-rows {0,1}, lanes 16-31 {2,3}.
// Base pointer centered at row 30: every row offset is a compile-time constant
// within the signed 24-bit IOFFSET range (|off| <= 6.2 MB < 8.39 MB), so each
// neighbor costs one 64-bit address computation + 32 immediate-offset loads.
__device__ __forceinline__ void load_Bfrags(const float* __restrict__ xb, int nb,
                                            int half, v2f Bf[16]) {
  const float* p = xb + (half * 2 + 30) * NB_N + nb;
#pragma unroll
  for (int cc = 0; cc < 16; ++cc) {
    Bf[cc].x = p[(cc * 4 - 30) * NB_N];
    Bf[cc].y = p[(cc * 4 - 29) * NB_N];
  }
}

// 64x16 += W[64-row slice] * B(64x16). Weights live in LDS in pair-packed form:
// element (d, c) at sWP[(c>>1)*PSTRIDE + d*2 + (c&1)], so each 16x4 A-fragment
// is ONE conflict-free ds_load_b64 per lane.
__device__ __forceinline__ void gemm_64x16(const float* __restrict__ sWP, int pbase,
                                           const v2f Bf[16], v8f acc[4],
                                           int half, int lid) {
#pragma unroll
  for (int cc = 0; cc < 16; ++cc) {
    const int p = pbase + cc * 2 + half;          // pair index = c0/2
#pragma unroll
    for (int dt = 0; dt < 4; ++dt) {
      v2f Af = *(const v2f*)(sWP + p * PSTRIDE + (dt * 16 + lid) * 2);
      acc[dt] = __builtin_amdgcn_wmma_f32_16x16x4_f32(
          false, Af, false, Bf[cc], (short)0, acc[dt], false, false);
    }
  }
}

__global__ __launch_bounds__(256) void edgeconv_fused_kernel(
    const float* __restrict__ x,      // [B, C, N]
    const int*   __restrict__ eidx,   // edge_index [2, B, N, K] int32; plane 0 used
    const float* __restrict__ W1,     // [C, C] row-major (d, c)
    const float* __restrict__ b1,     // [C]
    const float* __restrict__ W2,     // [OUT, 2C] row-major
    const float* __restrict__ b2,     // [OUT]
    float*       __restrict__ out)    // [B, OUT, N]
{
  __shared__ float sW1P[32 * PSTRIDE];            // 20 KB, pair-packed W1^T
  __shared__ float sW2P[64 * PSTRIDE];            // 40 KB, pair-packed W2^T
  __shared__ float sB1[64];
  __shared__ float sB2[64];
  __shared__ float sH[WAVES_PER_BLOCK][32 * 32];  // hmax staging [pair][node*2+j], 4 KB/wave

  const int tid  = threadIdx.x;
  const int lane = tid & 31;
  const int wave = tid >> 5;

  // ---- Stage weights into pair-packed LDS layout. LDS writes are consecutive
  // across the wave (conflict-free); global reads strided but L2-resident.
  for (int i = tid; i < 32 * 128; i += 256) {
    int p = i >> 7, t = i & 127;                  // t = d*2 + j, c = p*2 + j
    sW1P[p * PSTRIDE + t] = W1[(t >> 1) * 64 + p * 2 + (t & 1)];
  }
  for (int i = tid; i < 64 * 128; i += 256) {
    int p = i >> 7, t = i & 127;
    sW2P[p * PSTRIDE + t] = W2[(t >> 1) * 128 + p * 2 + (t & 1)];
  }
  if (tid < 64) { sB1[tid] = b1[tid]; sB2[tid] = b2[tid]; }
  __syncthreads();

  const int tile = blockIdx.x * WAVES_PER_BLOCK + wave;
  if (tile >= TILES_TOTAL) return;                // wave-uniform: EXEC stays all-ones

  const int half  = lane >> 4;                    // 0: K-rows {0,1}, 1: K-rows {2,3}
  const int lid   = lane & 15;                    // column (node) / A-matrix row index
  const int node0 = tile * 16;                    // flat (b*N + n) of first node in tile
  const int b     = node0 / NB_N;
  const int n_own = (node0 - b * NB_N) + lid;

  const float* xb = x + b * (NB_C * NB_N);

  // ---- b1 fragment in the 16x16 f32 C/D layout:
  //      VGPR v, lanes 0-15: (M=v, N=lane); lanes 16-31: (M=v+8, N=lane-16)
  v8f bias1[4];
#pragma unroll
  for (int dt = 0; dt < 4; ++dt)
#pragma unroll
    for (int v = 0; v < 8; ++v)
      bias1[dt][v] = sB1[dt * 16 + v + 8 * half];

  v8f hmax[4];
#pragma unroll
  for (int dt = 0; dt < 4; ++dt) hmax[dt] = (v8f)0.f;  // relu(h) >= 0 => 0-init exact

  // ---- Hoist all 9 neighbor indices: one in-flight batch of loads.
  const int kflat0 = (b * NB_N + n_own) * NB_K;   // edge_index[0] plane
  int nbidx[NB_K];
#pragma unroll
  for (int k = 0; k < NB_K; ++k) nbidx[k] = eidx[kflat0 + k];

  // ================= GEMM1 over K=9 neighbors, software-pipelined ==============
  // Double-buffered B-fragments: gathers for neighbor k+1 are issued before the
  // WMMAs consuming neighbor k, so waits stay incremental (loadcnt<=32) and
  // 32 gathers overlap 64 WMMAs each iteration. Full unroll => static ping-pong.
  v2f Bbuf[2][16];
  load_Bfrags(xb, nbidx[0], half, Bbuf[0]);
#pragma unroll
  for (int k = 0; k < NB_K; ++k) {
    if (k + 1 < NB_K) load_Bfrags(xb, nbidx[k + 1], half, Bbuf[(k + 1) & 1]);

    v8f acc[4];
#pragma unroll
    for (int dt = 0; dt < 4; ++dt) acc[dt] = bias1[dt];

    gemm_64x16(sW1P, 0, Bbuf[k & 1], acc, half, lid);

    // fused relu + running max (hmax >= 0 throughout)
#pragma unroll
    for (int dt = 0; dt < 4; ++dt)
#pragma unroll
      for (int v = 0; v < 8; ++v)
        hmax[dt][v] = fmaxf(hmax[dt][v], acc[dt][v]);
  }

  // ================= GEMM2: out = relu(W2a*x_own + W2b*hmax + b2) ==============
  // Issue the (coalesced) own-node gathers first ...
  v2f Bown[16];
  load_Bfrags(xb, n_own, half, Bown);

  // ... stage hmax to LDS pair-packed [pair][node*2+j] while they fly ...
  // (row d -> pair d>>1, j = d&1 = v&1; d>>1 = dt*8 + 4*half + (v>>1))
  float* Hs = sH[wave];                           // wave-private; DS ops in-order per wave
#pragma unroll
  for (int dt = 0; dt < 4; ++dt)
#pragma unroll
    for (int v = 0; v < 8; ++v)
      Hs[(dt * 8 + 4 * half + (v >> 1)) * 32 + lid * 2 + (v & 1)] = hmax[dt][v];

  // bias2 fragment built only now, so it isn't live across the hot loop.
  v8f acc2[4];
#pragma unroll
  for (int dt = 0; dt < 4; ++dt)
#pragma unroll
    for (int v = 0; v < 8; ++v)
      acc2[dt][v] = sB2[dt * 16 + v + 8 * half];

  // ... run the LDS-fed half (cat rows 64..127 -> W2b * hmax) to hide the
  // global latency of Bown; B-fragments are single ds_load_b64 each ...
#pragma unroll
  for (int cc = 0; cc < 16; ++cc) {
    const int p = cc * 2 + half;
    v2f Bf = *(const v2f*)(Hs + p * 32 + lid * 2);
#pragma unroll
    for (int dt = 0; dt < 4; ++dt) {
      v2f Af = *(const v2f*)(sW2P + (32 + p) * PSTRIDE + (dt * 16 + lid) * 2);
      acc2[dt] = __builtin_amdgcn_wmma_f32_16x16x4_f32(
          false, Af, false, Bf, (short)0, acc2[dt], false, false);
    }
  }
  // ... then the x half (cat rows 0..63 -> W2a * x_own).
  gemm_64x16(sW2P, 0, Bown, acc2, half, lid);

  // ---- ReLU + non-temporal store, out[b][d][n]. Same base-centering trick:
  // row offsets are compile-time immediates (|off| <= 5.6 MB); streamed once,
  // so bypass cache retention (NT) and keep L2 for the x gather working set.
  float* po = out + b * (NB_OUT * NB_N) + (8 * half + 28) * NB_N + n_own;
#pragma unroll
  for (int dt = 0; dt < 4; ++dt)
#pragma unroll
    for (int v = 0; v < 8; ++v)
      __builtin_nontemporal_store(fmaxf(acc2[dt][v], 0.f),
                                  &po[(dt * 16 + v - 28) * NB_N]);
}

extern "C" void kernel_launch(void* const* d_in, const int* in_sizes, int n_in,
                              void* d_out, int out_size, void* d_ws, size_t ws_size,
                              hipStream_t stream) {
  (void)in_sizes; (void)n_in; (void)out_size; (void)d_ws; (void)ws_size;
  const float* x    = (const float*)d_in[0];
  const int*   eidx = (const int*)  d_in[1];
  const float* W1   = (const float*)d_in[2];
  const float* b1   = (const float*)d_in[3];
  const float* W2   = (const float*)d_in[4];
  const float* b2   = (const float*)d_in[5];
  float* out = (float*)d_out;

  const int blocks = (TILES_TOTAL + WAVES_PER_BLOCK - 1) / WAVES_PER_BLOCK;  // 782
  edgeconv_fused_kernel<<<blocks, 256, 0, stream>>>(x, eidx, W1, b1, W2, b2, out);
}